// Som_23089744183527
// MI455X (gfx1250) — compile-verified
//
#include <hip/hip_runtime.h>
#include <stdint.h>

typedef __attribute__((ext_vector_type(2))) float v2f;
typedef __attribute__((ext_vector_type(4))) float v4f;
typedef __attribute__((ext_vector_type(8))) float v8f;

#define B_    128
#define D_    256
#define ROWS_ 64
#define COLS_ 64
#define M_    (ROWS_ * COLS_)   // 4096 neurons

// ---------------------------------------------------------------------------
// ws layout: [0..127]   u64 per-batch-row packed (dist_key<<32 | m) atomic-min
//            [+128 u64] 4096 floats of w_sq
// ---------------------------------------------------------------------------

__global__ void init_ws_kernel(unsigned long long* __restrict__ ws_min) {
    int i = threadIdx.x;
    if (i < B_) ws_min[i] = 0xFFFFFFFFFFFFFFFFull;
}

// One wave per neuron row: sum of squares of 256 floats.
__global__ __launch_bounds__(32) void wsq_kernel(const float* __restrict__ w,
                                                 float* __restrict__ w_sq) {
    int m = blockIdx.x;
    int l = threadIdx.x;
    const float* row = w + (size_t)m * D_;
    v4f a = *(const v4f*)(row + l * 4);
    v4f b = *(const v4f*)(row + 128 + l * 4);
    float s = a.x * a.x + a.y * a.y + a.z * a.z + a.w * a.w +
              b.x * b.x + b.y * b.y + b.z * b.z + b.w * b.w;
    #pragma unroll
    for (int sh = 16; sh >= 1; sh >>= 1) s += __shfl_xor(s, sh, 32);
    if (l == 0) w_sq[m] = s;
}

// One wave computes a 16(B) x 16(M) tile of x @ w^T via V_WMMA_F32_16X16X4_F32,
// converts to relative squared distance (w_sq - 2*dot; |x|^2 is constant per
// batch row so it cannot change the argmin), packs into monotonic u64 keys and
// atomically min-combines into ws_min[batch_row].
__global__ __launch_bounds__(32) void som_argmin_wmma(
        const float* __restrict__ x, const float* __restrict__ w,
        const float* __restrict__ w_sq, unsigned long long* __restrict__ ws_min) {
    const int m0   = blockIdx.x * 16;     // neuron tile base
    const int b0   = blockIdx.y * 16;     // batch tile base
    const int l    = threadIdx.x;
    const int half = l >> 4;              // 0: K={0,1}, 1: K={2,3} (A/B layout)
    const int lid  = l & 15;
    const int kb   = half * 2;

    const float* xrow = x + (size_t)(b0 + lid) * D_;  // A: M=lid row of x tile
    const float* wrow = w + (size_t)(m0 + lid) * D_;  // B: N=lid col = w row

    v8f c = {};
    #pragma unroll 4
    for (int k0 = 0; k0 < D_; k0 += 4) {
        v2f a = *(const v2f*)(xrow + k0 + kb);
        v2f b = *(const v2f*)(wrow + k0 + kb);
        // 8 args: (neg_a, A, neg_b, B, c_mod, C, reuse_a, reuse_b)
        c = __builtin_amdgcn_wmma_f32_16x16x4_f32(
                false, a, false, b, (short)0, c, false, false);
    }

    const float        wsq  = w_sq[m0 + lid];
    const unsigned int mcol = (unsigned int)(m0 + lid);

    // C layout: VGPR i, lanes 0-15 -> row b0+i (col=lid);
    //           lanes 16-31 -> row b0+i+8 (col=lid).
    #pragma unroll
    for (int i = 0; i < 8; ++i) {
        float d2 = wsq - 2.0f * c[i];
        unsigned int u = __float_as_uint(d2);
        u = (u & 0x80000000u) ? ~u : (u | 0x80000000u);  // monotonic float->uint
        unsigned long long key = ((unsigned long long)u << 32) | mcol;
        #pragma unroll
        for (int sh = 8; sh >= 1; sh >>= 1) {            // min within 16-lane half
            unsigned long long o = __shfl_xor(key, sh, 32);
            key = (o < key) ? o : key;
        }
        if (lid == 0) {
            int brow = b0 + i + half * 8;
            atomicMin(&ws_min[brow], key);
        }
    }
}

__global__ void finalize_bmus_kernel(const unsigned long long* __restrict__ ws_min,
                                     float* __restrict__ bmus) {
    int b = threadIdx.x;
    if (b < B_) {
        unsigned int idx = (unsigned int)(ws_min[b] & 0xFFFFFFFFull);
        bmus[b * 2 + 0] = (float)(idx / COLS_);
        bmus[b * 2 + 1] = (float)(idx % COLS_);
    }
}

// diffs[b][m][d] = x[b][d] - w[m][d]; 512 MB streamed out.
// float4 per thread, non-temporal store so the hot 4 MB `w` stays resident in L2.
__global__ __launch_bounds__(256) void diffs_kernel(const float* __restrict__ x,
                                                    const float* __restrict__ w,
                                                    float* __restrict__ diffs) {
    size_t tid = (size_t)blockIdx.x * blockDim.x + threadIdx.x;
    int    d4  = (int)(tid & 63);          // D_/4 = 64 float4 per (b,m) row
    size_t bm  = tid >> 6;
    int    m   = (int)(bm & (M_ - 1));
    int    b   = (int)(bm >> 12);          // M_ = 2^12
    v4f xv = *(const v4f*)(x + (size_t)b * D_ + (size_t)d4 * 4);
    v4f wv = *(const v4f*)(w + (size_t)m * D_ + (size_t)d4 * 4);
    v4f dv = xv - wv;
    __builtin_nontemporal_store(dv, (v4f*)(diffs + bm * D_ + (size_t)d4 * 4));
}

extern "C" void kernel_launch(void* const* d_in, const int* in_sizes, int n_in,
                              void* d_out, int out_size, void* d_ws, size_t ws_size,
                              hipStream_t stream) {
    (void)in_sizes; (void)n_in; (void)out_size; (void)ws_size;
    const float* x = (const float*)d_in[0];   // [128, 256]
    const float* w = (const float*)d_in[1];   // [64, 64, 256]

    float* out   = (float*)d_out;
    float* bmus  = out;            // [128, 2] flat (return order first)
    float* diffs = out + 2 * B_;   // [128, 64, 64, 256] flat

    unsigned long long* ws_min = (unsigned long long*)d_ws;
    float*              w_sq   = (float*)(ws_min + B_);

    // argmin pipeline (same stream => ordered)
    init_ws_kernel<<<1, 128, 0, stream>>>(ws_min);
    wsq_kernel<<<M_, 32, 0, stream>>>(w, w_sq);
    dim3 gtile(M_ / 16, B_ / 16);  // 256 x 8 waves
    som_argmin_wmma<<<gtile, 32, 0, stream>>>(x, w, w_sq, ws_min);
    finalize_bmus_kernel<<<1, 128, 0, stream>>>(ws_min, bmus);

    // bandwidth-bound diffs tensor
    size_t total4 = (size_t)B_ * M_ * (D_ / 4);   // 33,554,432 float4
    diffs_kernel<<<(unsigned)(total4 / 256), 256, 0, stream>>>(x, w, diffs);
}